// CRDLoss_v2_11295763988757
// MI455X (gfx1250) — compile-verified
//
#include <hip/hip_runtime.h>
#include <hip/hip_bf16.h>

typedef __attribute__((ext_vector_type(2))) float v2f;
typedef __attribute__((ext_vector_type(8))) float v8f;

#define BB   256      // batch
#define DD   128      // feature dim
#define KK1  4097     // 1 pos + 4096 neg
#define NDATA 500000
#define NCE_T 0.07f
#define MPN  (4096.0f / 500000.0f)   // m * Pn
#define EPSV 1e-7f

// ---- deterministic block tree-reduction (power-of-2 blockDim) ----
__device__ __forceinline__ float block_reduce_sum(float v, float* red) {
    const int t = threadIdx.x;
    red[t] = v;
    __syncthreads();
    for (int s = blockDim.x >> 1; s > 0; s >>= 1) {
        if (t < s) red[t] += red[t + s];
        __syncthreads();
    }
    float r = red[0];
    __syncthreads();
    return r;
}

// ---- Kernel 1: gathered dot products via V_WMMA_F32_16X16X4_F32 ----
// Block = one batch element b (256 thr = 8 waves). Wave handles k-tiles of 16
// rows. A = 16 gathered memory rows (16x4 f32 chunk), B = f_s[b] chunk
// broadcast across all 16 N columns, accumulate C over 32 chunks (D=128).
__global__ __launch_bounds__(256) void k_nce_dots(
    const float* __restrict__ f_s, const float* __restrict__ mem,
    const int* __restrict__ idx, const int* __restrict__ cidx,
    float* __restrict__ E, float* __restrict__ partZ)
{
    __shared__ float fs_sh[DD];
    __shared__ float red[256];
    const int b   = blockIdx.x;
    const int tid = threadIdx.x;

    // stage L2-normalized f_s[b] into LDS
    float v  = (tid < DD) ? f_s[b * DD + tid] : 0.0f;
    float ss = block_reduce_sum(v * v, red);
    float inv = 1.0f / sqrtf(ss);
    if (tid < DD) fs_sh[tid] = v * inv;
    __syncthreads();

    const int wave = tid >> 5;
    const int lane = tid & 31;
    const int m    = lane & 15;            // row within tile
    const int off  = (lane >> 4) << 1;     // lanes<16: K{0,1}; lanes>=16: K{2,3}

    float local = 0.0f;
    for (int t = wave; t < 257; t += 8) {  // 257 tiles of 16 cover 4097
        const int k = t * 16 + m;
        int r = 0;
        if (k < KK1) r = (k == 0) ? idx[b] : cidx[b * 4096 + (k - 1)];
        const float* __restrict__ row = mem + (size_t)r * DD;

        v8f acc = {};
        #pragma unroll 8
        for (int c = 0; c < DD / 4; ++c) {
            const int d = c * 4 + off;
            v2f a;  a.x  = row[d];     a.y  = row[d + 1];
            v2f bb; bb.x = fs_sh[d];   bb.y = fs_sh[d + 1];
            acc = __builtin_amdgcn_wmma_f32_16x16x4_f32(
                false, a, false, bb, (short)0, acc, false, false);
        }

        // N=0 column: lane 0 holds M=0..7, lane 16 holds M=8..15
        if ((lane & 15) == 0) {
            const int base = t * 16 + ((lane >> 4) << 3);
            float ls = 0.0f;
            #pragma unroll
            for (int j = 0; j < 8; ++j) {
                const int kk = base + j;
                if (kk < KK1) {
                    float e = expf(acc[j] * (1.0f / NCE_T));
                    E[(size_t)b * KK1 + kk] = e;
                    ls += e;
                }
            }
            local += ls;
        }
    }
    float tot = block_reduce_sum(local, red);
    if (tid == 0) partZ[b] = tot;
}

// ---- Kernel 2: Z = mean(out) * N_DATA ----
__global__ __launch_bounds__(256) void k_zreduce(
    const float* __restrict__ partZ, float* __restrict__ Zbuf)
{
    __shared__ float red[256];
    float s = block_reduce_sum(partZ[threadIdx.x], red);
    if (threadIdx.x == 0)
        Zbuf[0] = (s / ((float)BB * (float)KK1)) * (float)NDATA;
}

// ---- Kernel 3: per-element log terms, block partial sums ----
__global__ __launch_bounds__(256) void k_lossterms(
    const float* __restrict__ E, const float* __restrict__ Zbuf,
    float* __restrict__ partL)
{
    __shared__ float red[256];
    const float zinv = 1.0f / Zbuf[0];
    const int NE = BB * KK1;
    float local = 0.0f;
    for (int i = blockIdx.x * blockDim.x + threadIdx.x; i < NE;
         i += gridDim.x * blockDim.x) {
        const float o = E[i] * zinv;
        const int  k  = i % KK1;
        const float denom = o + MPN + EPSV;
        local += (k == 0) ? logf(o / denom) : logf(MPN / denom);
    }
    float s = block_reduce_sum(local, red);
    if (threadIdx.x == 0) partL[blockIdx.x] = s;
}

// ---- Kernel 4: final loss scalar ----
__global__ __launch_bounds__(256) void k_lossfinal(
    const float* __restrict__ partL, float* __restrict__ out)
{
    __shared__ float red[256];
    float s = block_reduce_sum(partL[threadIdx.x], red);
    if (threadIdx.x == 0) out[0] = -(s / (float)BB);  // P == 1
}

// ---- Kernel 5: new_memory = copy of memory_t (dst is 4B-aligned only) ----
__global__ __launch_bounds__(256) void k_copymem(
    const float4* __restrict__ src, float* __restrict__ dst)
{
    const size_t n4 = (size_t)NDATA * DD / 4;
    size_t i = (size_t)blockIdx.x * blockDim.x + threadIdx.x;
    const size_t stride = (size_t)gridDim.x * blockDim.x;
    for (; i < n4; i += stride) {
        float4 v = src[i];
        dst[4 * i + 0] = v.x;
        dst[4 * i + 1] = v.y;
        dst[4 * i + 2] = v.z;
        dst[4 * i + 3] = v.w;
    }
}

// ---- Kernel 6: overwrite rows idx with l2norm(0.5*mem[idx] + 0.5*l2norm(f_t)) ----
__global__ __launch_bounds__(128) void k_update(
    const float* __restrict__ f_t, const float* __restrict__ mem,
    const int* __restrict__ idx, float* __restrict__ out_mem)
{
    __shared__ float red[128];
    const int b = blockIdx.x;
    const int t = threadIdx.x;
    float fv  = f_t[b * DD + t];
    float ss1 = block_reduce_sum(fv * fv, red);
    float ftn = fv * (1.0f / sqrtf(ss1));
    const int r = idx[b];
    float w   = mem[(size_t)r * DD + t] * 0.5f + ftn * 0.5f;
    float ss2 = block_reduce_sum(w * w, red);
    out_mem[(size_t)r * DD + t] = w * (1.0f / sqrtf(ss2));
}

extern "C" void kernel_launch(void* const* d_in, const int* in_sizes, int n_in,
                              void* d_out, int out_size, void* d_ws, size_t ws_size,
                              hipStream_t stream) {
    const float* f_s  = (const float*)d_in[0];
    const float* f_t  = (const float*)d_in[1];
    const float* mem  = (const float*)d_in[2];
    const int*   idx  = (const int*)d_in[3];
    const int*   cidx = (const int*)d_in[4];
    // d_in[5] = epoch (unused)

    float* out     = (float*)d_out;
    float* out_mem = out + 1;                 // new_memory, flat after loss

    float* E     = (float*)d_ws;              // BB*KK1 exp values (~4.2 MB)
    float* partZ = E + (size_t)BB * KK1;      // 256
    float* Zbuf  = partZ + BB;                // 1
    float* partL = Zbuf + 1;                  // 256

    k_nce_dots <<<BB,   256, 0, stream>>>(f_s, mem, idx, cidx, E, partZ);
    k_zreduce  <<<1,    256, 0, stream>>>(partZ, Zbuf);
    k_lossterms<<<256,  256, 0, stream>>>(E, Zbuf, partL);
    k_lossfinal<<<1,    256, 0, stream>>>(partL, out);
    k_copymem  <<<4096, 256, 0, stream>>>((const float4*)mem, out_mem);
    k_update   <<<BB,   128, 0, stream>>>(f_t, mem, idx, out_mem);
}